// Net_73607149519409
// MI455X (gfx1250) — compile-verified
//
#include <hip/hip_runtime.h>
#include <math.h>

#define BATCH 8
#define NPTS  2048
#define KNN   32
#define NBINS 9
#define WIDTH_F 20.0f
#define RAD2DEG 57.29577951308232f
#define WAVES 4   // waves per block in knn_kernel

typedef __attribute__((ext_vector_type(2))) float v2f;
typedef __attribute__((ext_vector_type(8))) float v8f;

__device__ __forceinline__ float pmodf(float a, float m) {
    // python-style positive modulo
    return a - m * floorf(a / m);
}

// ---------------------------------------------------------------------------
// Kernel 1: fused KNN. 4 waves per block; each wave owns one 16-query tile.
// Gram tiles via V_WMMA_F32_16X16X4_F32 (K=3 padded to 4). Top-32 kept in an
// LDS min-tracked unsorted buffer. Points stored xyz0 (w==0) so WMMA fragment
// loads are single unconditional 8-byte LDS reads; squared norms in sq[].
// Never materializes the 2048x2048 distance matrix in HBM.
// ---------------------------------------------------------------------------
__global__ __launch_bounds__(32 * WAVES)
void knn_kernel(const float* __restrict__ x, int* __restrict__ idx_out) {
    __shared__ float pts[NPTS * 4];            // xyz0, 32KB (shared by all waves)
    __shared__ float sq[NPTS];                 // |p|^2, 8KB
    __shared__ float tile[WAVES][16 * 16];     // per-wave WMMA D tile, 4KB
    __shared__ float cs[WAVES][KNN * 32];      // per-wave candidate scores [slot][lane], 16KB
    __shared__ int   ci[WAVES][KNN * 32];      // per-wave candidate indices, 16KB

    const int tid  = threadIdx.x;
    const int lane = tid & 31;
    const int wave = tid >> 5;
    const int b    = blockIdx.y;
    const int q0   = blockIdx.x * (16 * WAVES) + wave * 16;

    // Load all points of this batch (x is (B,3,N)); w slot = 0 for K=4 padding.
    const float* xb = x + (size_t)b * 3 * NPTS;
    for (int i = tid; i < NPTS; i += 32 * WAVES) {
        float px = xb[i];
        float py = xb[NPTS + i];
        float pz = xb[2 * NPTS + i];
        pts[i * 4 + 0] = px;
        pts[i * 4 + 1] = py;
        pts[i * 4 + 2] = pz;
        pts[i * 4 + 3] = 0.0f;
        sq[i] = px * px + py * py + pz * pz;
    }

    // Init this lane's candidate buffer (unsorted, min-tracked).
    float* mycs = &cs[wave][0];
    int*   myci = &ci[wave][0];
#pragma unroll
    for (int j = 0; j < KNN; ++j) {
        mycs[j * 32 + lane] = -3.0e38f;
        myci[j * 32 + lane] = 0;
    }
    float curmin = -3.0e38f;
    int   minpos = 0;

    __syncthreads();

    const int   row  = lane >> 1;
    const int   half = lane & 1;
    const float qq   = sq[q0 + row];

    // WMMA fragment coordinates (ISA 7.12.2: 32-bit A 16x4 layout):
    // lane l holds row/col (l&15); K pair = {0,1} for lanes 0-15, {2,3} for 16-31.
    const int m  = lane & 15;
    const int kk = (lane < 16) ? 0 : 2;

    // Queries fragment: one aligned 8-byte LDS read (w slot is 0).
    v2f afrag = *(const v2f*)&pts[(q0 + m) * 4 + kk];

    for (int p0 = 0; p0 < NPTS; p0 += 16) {
        // Neighbor tile fragment (4x16, mirror striping): one ds_load_b64.
        v2f bfrag = *(const v2f*)&pts[(p0 + m) * 4 + kk];

        v8f c = {};
        // D = A x B : 16x16 tile of dot(query_m, point_n)
        c = __builtin_amdgcn_wmma_f32_16x16x4_f32(
                false, afrag, false, bfrag, (short)0, c, false, false);

        // Scatter D (lane-striped) -> LDS tile[M][N]
#pragma unroll
        for (int j = 0; j < 8; ++j) {
            int rm = (lane < 16) ? j : (j + 8);
            tile[wave][rm * 16 + m] = c[j];
        }
        __syncthreads();

        // Consume: lane owns (row, half) -> 8 neighbor columns
#pragma unroll
        for (int i = 0; i < 8; ++i) {
            int   n  = half * 8 + i;
            int   gi = p0 + n;
            float s  = 2.0f * tile[wave][row * 16 + n] - qq - sq[gi];
            if (s > curmin) {
                // replace current worst, then rescan for the new worst
                mycs[minpos * 32 + lane] = s;
                myci[minpos * 32 + lane] = gi;
                curmin = mycs[lane];
                minpos = 0;
                for (int j = 1; j < KNN; ++j) {
                    float v = mycs[j * 32 + lane];
                    if (v < curmin) { curmin = v; minpos = j; }
                }
            }
        }
        __syncthreads();
    }

    // Final: lanes 0..15 of each wave produce one sorted row each from the
    // 64 unsorted candidates of lanes (2r, 2r+1).
    if (lane < 16) {
        int r  = lane;
        int c0 = 2 * r;
        int* outp = idx_out + ((size_t)b * NPTS + q0 + r) * KNN;
        unsigned long long used = 0ull;
        for (int o = 0; o < KNN; ++o) {
            float best  = -3.0e38f;
            int   bestt = 0;
            int   besti = 0x7fffffff;
            for (int t = 0; t < 64; ++t) {
                if ((used >> t) & 1ull) continue;
                int col = c0 + (t >> 5);
                int j   = t & 31;
                float v = mycs[j * 32 + col];
                int  gi = myci[j * 32 + col];
                if (v > best || (v == best && gi < besti)) {
                    best = v; bestt = t; besti = gi;
                }
            }
            used |= (1ull << bestt);
            outp[o] = besti;
        }
    }
}

// ---------------------------------------------------------------------------
// Kernel 2: per-point 3x3 covariance of the 32-neighborhood, top eigenvector
// by power iteration (== first right singular vector of the centered 32x3),
// magnitude = lambda^(1/4) == sqrt(sigma0).
// ---------------------------------------------------------------------------
__global__ __launch_bounds__(256)
void eig_kernel(const float* __restrict__ x, const int* __restrict__ idx,
                float* __restrict__ grad, float* __restrict__ mag) {
    int pt = blockIdx.x * blockDim.x + threadIdx.x;
    if (pt >= BATCH * NPTS) return;
    int b = pt / NPTS;
    const float* xb = x + (size_t)b * 3 * NPTS;
    const int*   nb = idx + (size_t)pt * KNN;

    float sx = 0, sy = 0, sz = 0;
    float sxx = 0, syy = 0, szz = 0, sxy = 0, sxz = 0, syz = 0;
    for (int j = 0; j < KNN; ++j) {
        int   i  = nb[j];
        float px = xb[i], py = xb[NPTS + i], pz = xb[2 * NPTS + i];
        sx += px; sy += py; sz += pz;
        sxx += px * px; syy += py * py; szz += pz * pz;
        sxy += px * py; sxz += px * pz; syz += py * pz;
    }
    const float inv = 1.0f / (float)KNN;
    float mx = sx * inv, my = sy * inv, mz = sz * inv;
    float cxx = sxx - (float)KNN * mx * mx;
    float cyy = syy - (float)KNN * my * my;
    float czz = szz - (float)KNN * mz * mz;
    float cxy = sxy - (float)KNN * mx * my;
    float cxz = sxz - (float)KNN * mx * mz;
    float cyz = syz - (float)KNN * my * mz;

    float vx = 0.577350269f, vy = 0.577350269f, vz = 0.577350269f;
    float lam = 0.0f;
    for (int it = 0; it < 32; ++it) {
        float wx = cxx * vx + cxy * vy + cxz * vz;
        float wy = cxy * vx + cyy * vy + cyz * vz;
        float wz = cxz * vx + cyz * vy + czz * vz;
        float nrm = sqrtf(wx * wx + wy * wy + wz * wz);
        lam = nrm;
        float r = (nrm > 1e-20f) ? (1.0f / nrm) : 0.0f;
        vx = wx * r; vy = wy * r; vz = wz * r;
    }
    grad[(size_t)pt * 3 + 0] = vx;
    grad[(size_t)pt * 3 + 1] = vy;
    grad[(size_t)pt * 3 + 2] = vz;
    mag[pt] = sqrtf(sqrtf(fmaxf(lam, 0.0f)));  // sigma0^(1/2) = lambda^(1/4)
}

// ---------------------------------------------------------------------------
// Kernel 3: gather neighbor gradients/magnitudes, angle binning, soft votes,
// per-channel L2 normalization, write in the reference's flat reshape order.
// ---------------------------------------------------------------------------
__global__ __launch_bounds__(256)
void hog_kernel(const int* __restrict__ idx, const float* __restrict__ grad,
                const float* __restrict__ mag, float* __restrict__ out) {
    int pt = blockIdx.x * blockDim.x + threadIdx.x;
    if (pt >= BATCH * NPTS) return;
    int b = pt / NPTS;
    int n = pt - b * NPTS;
    const int* nb = idx + (size_t)pt * KNN;

    float hist[NBINS][2];
#pragma unroll
    for (int q = 0; q < NBINS; ++q) { hist[q][0] = 0.0f; hist[q][1] = 0.0f; }

    for (int j = 0; j < KNN; ++j) {
        int   g  = b * NPTS + nb[j];
        float gx = grad[(size_t)g * 3 + 0];
        float gy = grad[(size_t)g * 3 + 1];
        float gz = grad[(size_t)g * 3 + 2];
        float mm = mag[g];

        float ang[2];
        ang[0] = acosf(gz) * RAD2DEG;        // zenith
        ang[1] = atanf(gy / gx) * RAD2DEG;   // azimuth

#pragma unroll
        for (int c = 0; c < 2; ++c) {
            float a = (float)(int)ang[c];    // truncate toward zero (int cast)
            if (a < 0.0f) a += 180.0f;
            int ib = (int)floorf(a / WIDTH_F - 0.5f);
            ib = ((ib % NBINS) + NBINS) % NBINS;
            float fc = WIDTH_F * ((float)((ib + 1) % NBINS) + 0.5f);
            float fv = mm * pmodf(fc - a, 180.0f) / WIDTH_F;
            float sc = WIDTH_F * ((float)ib + 0.5f);
            float sv = mm * pmodf(a - sc, 180.0f) / WIDTH_F;
            hist[ib][c]               += fv;
            hist[(ib + 1) % NBINS][c] += sv;
        }
    }

#pragma unroll
    for (int c = 0; c < 2; ++c) {
        float s2 = 0.0f;
#pragma unroll
        for (int q = 0; q < NBINS; ++q) s2 += hist[q][c] * hist[q][c];
        float nrm = fmaxf(sqrtf(s2), 1e-12f);
#pragma unroll
        for (int q = 0; q < NBINS; ++q) hist[q][c] /= nrm;
    }

    // reference reshape: flat (n, q, c) order reinterpreted as (18, N)
    float* ob = out + (size_t)b * (2 * NBINS) * NPTS;
#pragma unroll
    for (int q = 0; q < NBINS; ++q) {
#pragma unroll
        for (int c = 0; c < 2; ++c) {
            ob[(size_t)n * (2 * NBINS) + q * 2 + c] = hist[q][c];
        }
    }
}

// ---------------------------------------------------------------------------
extern "C" void kernel_launch(void* const* d_in, const int* in_sizes, int n_in,
                              void* d_out, int out_size, void* d_ws, size_t ws_size,
                              hipStream_t stream) {
    (void)in_sizes; (void)n_in; (void)out_size; (void)ws_size;

    const float* x  = (const float*)d_in[0];  // (B,3,N) float32; d_in[1] = k (==32)
    float* out = (float*)d_out;

    // workspace layout: idx (2MB) | grad (192KB) | mag (64KB)
    char*  ws   = (char*)d_ws;
    int*   idx  = (int*)ws;
    float* grad = (float*)(ws + (size_t)BATCH * NPTS * KNN * sizeof(int));
    float* mag  = grad + (size_t)BATCH * NPTS * 3;

    dim3 g1(NPTS / (16 * WAVES), BATCH);
    knn_kernel<<<g1, 32 * WAVES, 0, stream>>>(x, idx);

    int total = BATCH * NPTS;
    eig_kernel<<<(total + 255) / 256, 256, 0, stream>>>(x, idx, grad, mag);
    hog_kernel<<<(total + 255) / 256, 256, 0, stream>>>(idx, grad, mag, out);
}